// MWERLoss_44495861187040
// MI455X (gfx1250) — compile-verified
//
#include <hip/hip_runtime.h>
#include <stdint.h>

// ---------------------------------------------------------------------------
// MWER loss for MI455X (gfx1250, wave32).
//   K1: TDM-stage emissions[b] into LDS, Gumbel-argmax sampling, CTC collapse
//   K2: path log-prob reduction via V_WMMA_F32_16X16X4_F32 (ones-matmul)
//   K3: exact Levenshtein DP, wave-per-problem, min-plus prefix via shuffles
//   K4: per-utterance softmax + expected WER + mean
// ---------------------------------------------------------------------------

#define NB 8
#define NT 500
#define NV 128
#define NL 100
#define NP 100
#define NBP (NB * NP)
#define WAVES_PER_BLOCK 8
#define BLOCKS_PER_B ((NP + WAVES_PER_BLOCK - 1) / WAVES_PER_BLOCK) // 13
#define NBEST_SCALE 0.5f

typedef __attribute__((ext_vector_type(2))) float v2f;
typedef __attribute__((ext_vector_type(8))) float v8f;
typedef __attribute__((ext_vector_type(4))) unsigned int v4u;
typedef __attribute__((ext_vector_type(8))) int v8i;
typedef __attribute__((ext_vector_type(4))) int v4i;

// Counter-based hash -> Gumbel(0,1) noise (deterministic per element).
__device__ __forceinline__ float gumbel_noise(unsigned idx) {
  unsigned h = idx * 747796405u + 2891336453u;
  h = ((h >> ((h >> 28) + 4u)) ^ h) * 277803737u;
  h = (h >> 22) ^ h;
  float u = ((float)(h >> 8) + 0.5f) * (1.0f / 16777216.0f); // (0,1)
  u = fminf(fmaxf(u, 1e-6f), 1.0f - 1e-6f);
  return -__logf(-__logf(u));
}

// Single-descriptor TDM: 1-D tile of `nelem` 4-byte elements, global -> LDS.
// D# layout per CDNA5 ISA (group0 128b, group1 256b).
// This toolchain exposes the 6-arg builtin:
//   (uint32x4 g0, int32x8 g1, int32x4 g2, int32x4 g3, int32x8 g4, i32 cpol)
__device__ __forceinline__ void tdm_load_f32_1d(const float* gsrc,
                                                float* lds_dst,
                                                unsigned nelem) {
  unsigned lds = (unsigned)(uintptr_t)(void*)lds_dst; // flat LDS addr[31:0] = LDS offset
  unsigned long long ga = (unsigned long long)(uintptr_t)gsrc;
  v4u g0;
  g0[0] = 1u;                                   // count=1 (valid user descriptor)
  g0[1] = lds;                                  // lds_addr
  g0[2] = (unsigned)ga;                         // global_addr[31:0]
  g0[3] = (unsigned)((ga >> 32) & 0x01ffffffu)  // global_addr[56:32]
          | 0x80000000u;                        // type=2 ("image") bits[127:126]=10b
  v8i g1;
  g1[0] = (int)(2u << 16);                      // workgroup_mask=0, data_size=2 (4B)
  g1[1] = (int)((nelem & 0xffffu) << 16);       // tensor_dim0[15:0]  @ bits 63:48
  g1[2] = (int)(((nelem >> 16) & 0xffffu)       // tensor_dim0[31:16] @ bits 79:64
                | (1u << 16));                  // tensor_dim1=1      @ bits 95:80
  g1[3] = (int)((nelem & 0xffffu) << 16);       // tile_dim0 (<=65535)@ bits 127:112
  g1[4] = 1;                                    // tile_dim1=1, tile_dim2=0
  g1[5] = (int)nelem;                           // tensor_dim0_stride[31:0]
  g1[6] = 0;                                    // stride hi / dim1_stride lo
  g1[7] = 0;
  v4i z4 = {0, 0, 0, 0};
  v8i z8 = {0, 0, 0, 0, 0, 0, 0, 0};
  __builtin_amdgcn_tensor_load_to_lds(g0, g1, z4, z4, z8, 0);
}

// ---------------------------------------------------------------------------
// K1: sample N-best paths + CTC collapse.
// Block = 8 waves (256 threads), one wave per path, 13 blocks per batch row.
// ---------------------------------------------------------------------------
__global__ void k_sample_collapse(const float* __restrict__ em,
                                  const int* __restrict__ em_len,
                                  int* __restrict__ hyp,
                                  int* __restrict__ hyp_len,
                                  float* __restrict__ lp_buf) {
  extern __shared__ float smem[];
  float* s_em = smem;                                       // NT*NV floats (250 KB)
  float* s_lp = smem + NT * NV;                             // 8*512 floats
  unsigned char* s_tok = (unsigned char*)(s_lp + WAVES_PER_BLOCK * 512);

  const int b = blockIdx.x / BLOCKS_PER_B;
  const int pblk = blockIdx.x % BLOCKS_PER_B;
  const int wave = (int)(threadIdx.x >> 5);
  const int lane = (int)(threadIdx.x & 31);
  const int p = pblk * WAVES_PER_BLOCK + wave;

  // Stage emissions[b] (T x V fp32) into LDS with one Tensor-DMA descriptor.
  if (wave == 0) {
    tdm_load_f32_1d(em + (size_t)b * NT * NV, s_em, (unsigned)(NT * NV));
    __builtin_amdgcn_s_wait_tensorcnt(0);
  }
  __syncthreads();

  const int tlen = em_len[b];

  if (p < NP) {
    const int bp = b * NP + p;
    float* w_lp = s_lp + wave * 512;
    unsigned char* w_tok = s_tok + wave * 512;
    const unsigned idx_base = (unsigned)bp * (unsigned)(NT * NV);

    for (int t = 0; t < NT; ++t) {
      // lane owns vocab [4*lane, 4*lane+3]; 128-bit LDS read, conflict-free.
      const float4 rv = ((const float4*)(s_em + t * NV))[lane];
      const unsigned gb = idx_base + (unsigned)t * (unsigned)NV + (unsigned)(lane * 4);
      float best = rv.x * NBEST_SCALE + gumbel_noise(gb + 0u);
      int bi = lane * 4;
      float v1 = rv.y * NBEST_SCALE + gumbel_noise(gb + 1u);
      float v2 = rv.z * NBEST_SCALE + gumbel_noise(gb + 2u);
      float v3 = rv.w * NBEST_SCALE + gumbel_noise(gb + 3u);
      if (v1 > best) { best = v1; bi = lane * 4 + 1; }
      if (v2 > best) { best = v2; bi = lane * 4 + 2; }
      if (v3 > best) { best = v3; bi = lane * 4 + 3; }
      // wave32 argmax reduce, first-index tie-break (matches jnp.argmax)
      #pragma unroll
      for (int off = 16; off; off >>= 1) {
        float ob = __shfl_xor(best, off, 32);
        int oi = __shfl_xor(bi, off, 32);
        if (ob > best || (ob == best && oi < bi)) { best = ob; bi = oi; }
      }
      if (lane == 0) {
        w_tok[t] = (unsigned char)bi;
        w_lp[t] = (t < tlen) ? s_em[t * NV + bi] : 0.0f; // frame-masked logprob
      }
    }
  }
  __syncthreads(); // make per-wave LDS scratch visible / ordered

  if (p < NP) {
    const int bp = b * NP + p;
    float* w_lp = s_lp + wave * 512;
    unsigned char* w_tok = s_tok + wave * 512;

    // coalesced write of masked per-frame logprobs (consumed by WMMA kernel)
    for (int c = 0; c < 16; ++c) {
      int t = c * 32 + lane;
      if (t < NT) lp_buf[(size_t)bp * NT + t] = w_lp[t];
    }

    // CTC collapse (drop blanks + repeats) via wave32 ballot compaction
    int base = 0;
    int prevtok = -1;
    int* hrow = hyp + (size_t)bp * NT;
    for (int c = 0; c < 16; ++c) {
      int t = c * 32 + lane;
      int tok = (t < NT) ? (int)w_tok[t] : 0;
      int up = __shfl_up(tok, 1, 32);
      int pv = (lane == 0) ? prevtok : up;
      bool keep = (t < tlen) && (tok != 0) && (tok != pv);
      unsigned mask = __builtin_amdgcn_ballot_w32(keep);
      int pos = base + __popc(mask & ((1u << lane) - 1u));
      if (keep) hrow[pos] = tok;
      base += __popc(mask);
      prevtok = __shfl(tok, 31, 32);
    }
    if (lane == 0) hyp_len[bp] = base;
  }
}

// ---------------------------------------------------------------------------
// K2: path_logp[p] = sum_t lp[p,t] via V_WMMA_F32_16X16X4_F32.
// One wave handles 16 paths; A = lp(16x4), B = ones(4x16), chained C accum.
// ---------------------------------------------------------------------------
__global__ void k_logp_wmma(const float* __restrict__ lp_buf,
                            float* __restrict__ path_logp) {
  const int lane = (int)(threadIdx.x & 31);
  const int base = blockIdx.x * 16;
  const int m = lane & 15;               // A-matrix row (path)
  const int khalf = (lane >> 4) * 2;     // lanes 0-15: K=0,1; lanes 16-31: K=2,3
  const float* rowp = lp_buf + (size_t)(base + m) * NT;

  v8f acc = {};
  v2f bones; bones[0] = 1.0f; bones[1] = 1.0f;
  for (int t = 0; t < NT; t += 4) {      // 500 = 125 * 4, exact
    v2f a;
    a[0] = rowp[t + khalf];
    a[1] = rowp[t + khalf + 1];
    acc = __builtin_amdgcn_wmma_f32_16x16x4_f32(
        /*neg_a=*/false, a, /*neg_b=*/false, bones,
        /*c_mod=*/(short)0, acc, /*reuse_a=*/false, /*reuse_b=*/false);
  }
  // C layout: VGPR j, lane 0 -> row j; VGPR j, lane 16 -> row 8+j (N=0 column)
  if (lane == 0) {
    #pragma unroll
    for (int j = 0; j < 8; ++j) path_logp[base + j] = acc[j];
  } else if (lane == 16) {
    #pragma unroll
    for (int j = 0; j < 8; ++j) path_logp[base + 8 + j] = acc[j];
  }
}

// ---------------------------------------------------------------------------
// K3: exact Levenshtein DP. Wave per (b,p); lane owns 4 contiguous columns.
// Row update: cand[j] = min(dp[j]+1, dp[j-1]+cost); then min-plus prefix
// new[j] = cummin(cand - j) + j  (insertion chain), matching the reference.
// ---------------------------------------------------------------------------
__global__ void k_edit(const int* __restrict__ hyp,
                       const int* __restrict__ hyp_len,
                       const int* __restrict__ labels,
                       const int* __restrict__ labels_len,
                       float* __restrict__ wers) {
  const int bp = (int)blockIdx.x;
  const int lane = (int)(threadIdx.x & 31);
  const int b = bp / NP;

  int refv[4], dp[4];
  #pragma unroll
  for (int k = 0; k < 4; ++k) {
    int j = lane * 4 + k;
    refv[k] = (j >= 1 && j <= NL) ? labels[b * NL + (j - 1)] : -999;
    dp[k] = j;
  }

  const int n = hyp_len[bp];
  const int* hrow = hyp + (size_t)bp * NT;
  const int BIG = 0x3fffffff;

  for (int i = 0; i < n; ++i) {
    int h = hrow[i];
    int dpl = __shfl_up(dp[3], 1, 32); // dp[j-1] across lane boundary
    int cand[4];
    #pragma unroll
    for (int k = 0; k < 4; ++k) {
      int left = (k == 0) ? dpl : dp[k - 1];
      int cost = (refv[k] != h) ? 1 : 0;
      int c = min(dp[k] + 1, left + cost);
      if (lane == 0 && k == 0) c = dp[0] + 1; // column 0: deletion count
      cand[k] = c;
    }
    // e = cand - j; intra-lane serial cummin
    int cm[4];
    int run = cand[0] - (lane * 4);
    cm[0] = run;
    #pragma unroll
    for (int k = 1; k < 4; ++k) {
      int e = cand[k] - (lane * 4 + k);
      run = min(run, e);
      cm[k] = run;
    }
    // cross-lane inclusive min-scan of lane carries, then shift to exclusive
    int x = run;
    #pragma unroll
    for (int off = 1; off < 32; off <<= 1) {
      int y = __shfl_up(x, off, 32);
      if (lane >= off) x = min(x, y);
    }
    int xs = __shfl_up(x, 1, 32);
    int excl = (lane == 0) ? BIG : xs;
    #pragma unroll
    for (int k = 0; k < 4; ++k) dp[k] = min(cm[k], excl) + (lane * 4 + k);
  }

  int rl = labels_len[b];
  int kk = rl & 3;
  int sel = dp[0];
  if (kk == 1) sel = dp[1];
  else if (kk == 2) sel = dp[2];
  else if (kk == 3) sel = dp[3];
  int ans = __shfl(sel, rl >> 2, 32);
  if (lane == 0) wers[bp] = (float)ans;
}

// ---------------------------------------------------------------------------
// K4: softmax over P per utterance, expected WER, mean over (B,P).
// ---------------------------------------------------------------------------
__global__ void k_loss(const float* __restrict__ path_logp,
                       const float* __restrict__ wers,
                       float* __restrict__ out) {
  __shared__ float s_acc[NB];
  const int wave = (int)(threadIdx.x >> 5); // = batch row b
  const int lane = (int)(threadIdx.x & 31);
  const float* lpb = path_logp + wave * NP;
  const float* wb = wers + wave * NP;

  float mx = -3.402823466e38f;
  for (int q = lane; q < NP; q += 32) mx = fmaxf(mx, lpb[q]);
  #pragma unroll
  for (int off = 16; off; off >>= 1) mx = fmaxf(mx, __shfl_xor(mx, off, 32));

  float se = 0.0f, sw = 0.0f;
  for (int q = lane; q < NP; q += 32) {
    float e = expf(lpb[q] - mx);
    se += e;
    sw += e * wb[q];
  }
  #pragma unroll
  for (int off = 16; off; off >>= 1) {
    se += __shfl_xor(se, off, 32);
    sw += __shfl_xor(sw, off, 32);
  }
  if (lane == 0) s_acc[wave] = sw / se; // E_b[WER]
  __syncthreads();
  if (threadIdx.x == 0) {
    float tot = 0.0f;
    #pragma unroll
    for (int i = 0; i < NB; ++i) tot += s_acc[i];
    out[0] = tot / (float)(NB * NP); // jnp.mean over (B,P)
  }
}

// ---------------------------------------------------------------------------
extern "C" void kernel_launch(void* const* d_in, const int* in_sizes, int n_in,
                              void* d_out, int out_size, void* d_ws,
                              size_t ws_size, hipStream_t stream) {
  const float* em = (const float*)d_in[0];      // (B,T,V) fp32
  const int* em_len = (const int*)d_in[1];      // (B,)
  const int* labels = (const int*)d_in[2];      // (B,L)
  const int* lab_len = (const int*)d_in[3];     // (B,)

  char* ws = (char*)d_ws;
  float* lp_buf = (float*)ws;    ws += (size_t)NBP * NT * sizeof(float);
  int* hyp = (int*)ws;           ws += (size_t)NBP * NT * sizeof(int);
  int* hyp_len = (int*)ws;       ws += (size_t)NBP * sizeof(int);
  float* path_logp = (float*)ws; ws += (size_t)NBP * sizeof(float);
  float* wers = (float*)ws;      ws += (size_t)NBP * sizeof(float);

  const size_t shmem = (size_t)NT * NV * sizeof(float)          // emissions tile
                       + WAVES_PER_BLOCK * 512 * sizeof(float)  // per-wave lp
                       + WAVES_PER_BLOCK * 512;                 // per-wave tokens

  k_sample_collapse<<<NB * BLOCKS_PER_B, WAVES_PER_BLOCK * 32, shmem, stream>>>(
      em, em_len, hyp, hyp_len, lp_buf);
  k_logp_wmma<<<NBP / 16, 32, 0, stream>>>(lp_buf, path_logp);
  k_edit<<<NBP, 32, 0, stream>>>(hyp, hyp_len, labels, lab_len, wers);
  k_loss<<<1, NB * 32, 0, stream>>>(path_logp, wers, (float*)d_out);
}